// CombinedTargetIOULoss_67929202754306
// MI455X (gfx1250) — compile-verified
//
#include <hip/hip_runtime.h>
#include <hip/hip_bf16.h>

// Problem constants from the reference
#define B_   64
#define K_   17
#define H_   64
#define W_   64
#define C_   (3 * K_)
#define P_   (H_ * W_)
#define EPS_ 1e-7f

typedef __attribute__((ext_vector_type(2))) float v2f;
typedef __attribute__((ext_vector_type(4))) float v4f;
typedef __attribute__((ext_vector_type(8))) float v8f;

__device__ __forceinline__ float frcp(float x) { return __builtin_amdgcn_rcpf(x); }

// Wave32 full-lane sum via V_WMMA_F32_16X16X4_F32.
// A layout (32-bit, 16x4): VGPR0 lane m<16 -> A[m][0], lane 16+m -> A[m][2];
// VGPR1 -> K=1/K=3 (we zero it). B = all ones (4x16).
// D[m][n] = v[m] + v[m+16] for every n. Summing the 8 D VGPRs per lane gives
// lanes<16: sum(v[0..7]) + sum(v[16..23]); lanes>=16: sum(v[8..15]) + sum(v[24..31]).
// One xor-16 shuffle completes the total (broadcast to all lanes).
__device__ __forceinline__ float wave_sum_wmma(float v) {
  v2f a;    a[0] = v;    a[1] = 0.0f;
  v2f ones; ones[0] = 1.0f; ones[1] = 1.0f;
  v8f c = {};
  c = __builtin_amdgcn_wmma_f32_16x16x4_f32(
        /*neg_a=*/false, a, /*neg_b=*/false, ones,
        /*c_mod=*/(short)0, c, /*reuse_a=*/false, /*reuse_b=*/false);
  float s = c[0] + c[1] + c[2] + c[3] + c[4] + c[5] + c[6] + c[7];
  s += __shfl_xor(s, 16, 32);
  return s;
}

// One block per (k, b) pair: streams P_ elements of 6 channels, accumulates
// {sum (tw*(hm_p-hm_g))^2, sum giou_loss*mask, sum mask} for that (k, b).
__global__ __launch_bounds__(256)
void ctl_partial_kernel(const float* __restrict__ outp,
                        const float* __restrict__ tgtp,
                        const float* __restrict__ twp,
                        float* __restrict__ part) {
  const int blk = blockIdx.x;          // blk = k * B_ + b
  const int k   = blk / B_;
  const int b   = blk % B_;
  const size_t base = ((size_t)b * C_ + (size_t)3 * k) * P_;

  const v4f* hp = (const v4f*)(outp + base);
  const v4f* xp = (const v4f*)(outp + base + P_);
  const v4f* yp = (const v4f*)(outp + base + 2 * P_);
  const v4f* hg = (const v4f*)(tgtp + base);
  const v4f* xg = (const v4f*)(tgtp + base + P_);
  const v4f* yg = (const v4f*)(tgtp + base + 2 * P_);
  const float tw = twp[b * K_ + k];    // uniform -> scalar load

  float mse_acc = 0.0f, num_acc = 0.0f, den_acc = 0.0f;

#pragma unroll
  for (int i = 0; i < 4; ++i) {
    const int j = threadIdx.x + i * 256;   // 1024 float4 per channel
    v4f hmp = __builtin_nontemporal_load(hp + j);
    v4f oxp = __builtin_nontemporal_load(xp + j);
    v4f oyp = __builtin_nontemporal_load(yp + j);
    v4f hmg = __builtin_nontemporal_load(hg + j);
    v4f oxg = __builtin_nontemporal_load(xg + j);
    v4f oyg = __builtin_nontemporal_load(yg + j);

#pragma unroll
    for (int e = 0; e < 4; ++e) {
      // MSE part (heatmap channel scaled by target weight)
      float hps = hmp[e] * tw;
      float hgs = hmg[e] * tw;
      float d = hps - hgs;
      mse_acc = fmaf(d, d, mse_acc);

      // GIoU part: xs/ys cancel -> everything in terms of min/max(offset, 0)
      float ax1 = fminf(oxp[e], 0.0f), ax2 = fmaxf(oxp[e], 0.0f);
      float ay1 = fminf(oyp[e], 0.0f), ay2 = fmaxf(oyp[e], 0.0f);
      float bx1 = fminf(oxg[e], 0.0f), bx2 = fmaxf(oxg[e], 0.0f);
      float by1 = fminf(oyg[e], 0.0f), by2 = fmaxf(oyg[e], 0.0f);

      float iw = fmaxf(fminf(ax2, bx2) - fmaxf(ax1, bx1), 0.0f);
      float ih = fmaxf(fminf(ay2, by2) - fmaxf(ay1, by1), 0.0f);
      float inter  = iw * ih;
      float area_p = (ax2 - ax1) * (ay2 - ay1);
      float area_g = (bx2 - bx1) * (by2 - by1);
      float uni    = area_p + area_g - inter + EPS_;
      float cw = fmaxf(ax2, bx2) - fminf(ax1, bx1);
      float ch = fmaxf(ay2, by2) - fminf(ay1, by1);
      float area_c = cw * ch + EPS_;

      // 1 - giou = 2 - inter/uni - uni/area_c
      //          = 2 - (inter*area_c + uni^2) / (uni*area_c)   -> single rcp
      float gl = 2.0f - fmaf(inter, area_c, uni * uni) * frcp(uni * area_c);

      float mask = (hgs != 0.0f) ? 1.0f : 0.0f;
      num_acc = fmaf(gl, mask, num_acc);
      den_acc += mask;
    }
  }

  // Wave-level reduction via WMMA, then cross-wave via LDS (deterministic).
  __shared__ float red[8][3];
  const int wave = threadIdx.x >> 5;
  const int lane = threadIdx.x & 31;

  float sm = wave_sum_wmma(mse_acc);
  float sn = wave_sum_wmma(num_acc);
  float sd = wave_sum_wmma(den_acc);

  if (lane == 0) { red[wave][0] = sm; red[wave][1] = sn; red[wave][2] = sd; }
  __syncthreads();

  if (threadIdx.x == 0) {
    float a0 = 0.0f, a1 = 0.0f, a2 = 0.0f;
#pragma unroll
    for (int i = 0; i < 8; ++i) { a0 += red[i][0]; a1 += red[i][1]; a2 += red[i][2]; }
    float* o = part + (size_t)blk * 3;
    o[0] = a0; o[1] = a1; o[2] = a2;
  }
}

// Single wave: per-k reduction over B partials (f64 accum), then wave sum.
__global__ __launch_bounds__(32)
void ctl_finalize_kernel(const float* __restrict__ part, float* __restrict__ out) {
  const int t = threadIdx.x;
  float contrib = 0.0f;
  if (t < K_) {
    double ms = 0.0, nm = 0.0, dn = 0.0;
    for (int b = 0; b < B_; ++b) {
      const float* p = part + ((size_t)t * B_ + b) * 3;
      ms += (double)p[0];
      nm += (double)p[1];
      dn += (double)p[2];
    }
    double mse = 0.5 * ms / (double)((size_t)B_ * P_);
    double den = dn < 1.0 ? 1.0 : dn;
    contrib = (float)(mse + nm / den);
  }
#pragma unroll
  for (int off = 16; off > 0; off >>= 1) contrib += __shfl_xor(contrib, off, 32);
  if (t == 0) out[0] = contrib / (float)K_;   // LOSS_WEIGHT == 1.0
}

extern "C" void kernel_launch(void* const* d_in, const int* in_sizes, int n_in,
                              void* d_out, int out_size, void* d_ws, size_t ws_size,
                              hipStream_t stream) {
  const float* outp = (const float*)d_in[0];   // output  (B, C, H, W) f32
  const float* tgtp = (const float*)d_in[1];   // target  (B, C, H, W) f32
  const float* twp  = (const float*)d_in[2];   // target_weights (B, K) f32
  float* out  = (float*)d_out;                 // scalar loss
  float* part = (float*)d_ws;                  // K_*B_*3 floats of partials

  ctl_partial_kernel<<<B_ * K_, 256, 0, stream>>>(outp, tgtp, twp, part);
  ctl_finalize_kernel<<<1, 32, 0, stream>>>(part, out);
}